// BahdanauAttention_26061861552323
// MI455X (gfx1250) — compile-verified
//
#include <hip/hip_runtime.h>
#include <math.h>

typedef __attribute__((ext_vector_type(2))) float v2f;
typedef __attribute__((ext_vector_type(4))) float f4;   // vector type usable with __builtin_nontemporal_load
typedef __attribute__((ext_vector_type(8))) float v8f;

#define B_ 64
#define S_ 1024
#define U_ 512
#define H_ 512
#define D_ 512

// ---------------------------------------------------------------------------
// Kernel 1: dec_attn[B,U] = dec_state[B,D] @ W_dec[D,U] + b_dec
// One wave (32 lanes) per 16x16 output tile, V_WMMA_F32_16X16X4_F32 over K.
// A 16x4 layout: lanes 0-15 -> M=lane, VGPR{0,1}=K{k,k+1}; lanes 16-31 -> K{k+2,k+3}
// B 4x16 layout: lanes 0-15 -> N=lane, VGPR{0,1}=K{k,k+1}; lanes 16-31 -> K{k+2,k+3}
// C/D 16x16:     VGPR r -> row m0+r (lanes 0-15) / m0+8+r (lanes 16-31), col n0+(lane&15)
// ---------------------------------------------------------------------------
__global__ __launch_bounds__(32) void k_dec_gemm(const float* __restrict__ dec_state,
                                                 const float* __restrict__ W_dec,
                                                 const float* __restrict__ b_dec,
                                                 float* __restrict__ dec_attn) {
  const int lane = threadIdx.x;
  const int m0   = blockIdx.x * 16;   // batch-row tile
  const int n0   = blockIdx.y * 16;   // U-col tile
  const int half = lane >> 4;         // 0: K pair {k,k+1}, 1: {k+2,k+3}
  const int l16  = lane & 15;

  v8f acc = {};
  for (int k = 0; k < D_; k += 4) {
    v2f a, b;
    const float* arow = dec_state + (m0 + l16) * D_ + k + half * 2;
    a.x = arow[0];
    a.y = arow[1];
    const float* bcol = W_dec + (k + half * 2) * U_ + n0 + l16;
    b.x = bcol[0];
    b.y = bcol[U_];
    acc = __builtin_amdgcn_wmma_f32_16x16x4_f32(false, a, false, b,
                                                (short)0, acc, false, false);
  }

  const int   n    = n0 + l16;
  const float bias = b_dec[n];
#pragma unroll
  for (int r = 0; r < 8; ++r) {
    const int m = m0 + r + half * 8;
    dec_attn[m * U_ + n] = acc[r] + bias;
  }
}

// ---------------------------------------------------------------------------
// Kernel 2: scores[b,s] = sum_u tanh(enc_attn + cov*W_cov + dec_attn) * w_attn
// One wave per (b,s) row; NT float4 streaming of enc_attn (single-use, 134 MB).
// ---------------------------------------------------------------------------
__global__ __launch_bounds__(256) void k_scores(const float* __restrict__ enc_attn,
                                                const float* __restrict__ coverage,
                                                const float* __restrict__ dec_attn,
                                                const float* __restrict__ W_cov,
                                                const float* __restrict__ w_attn,
                                                float* __restrict__ scores) {
  const int wave = threadIdx.x >> 5;
  const int lane = threadIdx.x & 31;
  const int row  = blockIdx.x * 8 + wave;   // row in [0, B*S)
  const int b    = row >> 10;               // row / S_

  const float cov = coverage[row];
  const f4* ea = (const f4*)(enc_attn + (size_t)row * U_);
  const f4* da = (const f4*)(dec_attn + (size_t)b * U_);
  const f4* wc = (const f4*)W_cov;
  const f4* wa = (const f4*)w_attn;

  float part = 0.f;
#pragma unroll
  for (int i = 0; i < 4; ++i) {
    const int j = i * 32 + lane;            // 128 float4 = 512 floats per wave
    const f4 e = __builtin_nontemporal_load(ea + j);  // streamed once: NT hint
    const f4 d = da[j];
    const f4 c = wc[j];
    const f4 w = wa[j];
    part += tanhf(e.x + cov * c.x + d.x) * w.x;
    part += tanhf(e.y + cov * c.y + d.y) * w.y;
    part += tanhf(e.z + cov * c.z + d.z) * w.z;
    part += tanhf(e.w + cov * c.w + d.w) * w.w;
  }
#pragma unroll
  for (int off = 16; off > 0; off >>= 1) part += __shfl_xor(part, off, 32);
  if (lane == 0) scores[row] = part;
}

// ---------------------------------------------------------------------------
// Kernel 3: softmax over S per batch row; writes attn to d_out.
// ---------------------------------------------------------------------------
__global__ __launch_bounds__(1024) void k_softmax(const float* __restrict__ scores,
                                                  float* __restrict__ attn) {
  __shared__ float red[32];
  const int b    = blockIdx.x;
  const int tid  = threadIdx.x;
  const int lane = tid & 31;
  const int wave = tid >> 5;

  const float v = scores[b * S_ + tid];

  // -------- max reduce --------
  float m = v;
#pragma unroll
  for (int off = 16; off > 0; off >>= 1) m = fmaxf(m, __shfl_xor(m, off, 32));
  if (lane == 0) red[wave] = m;
  __syncthreads();
  if (wave == 0) {
    float t = red[lane];
#pragma unroll
    for (int off = 16; off > 0; off >>= 1) t = fmaxf(t, __shfl_xor(t, off, 32));
    if (lane == 0) red[0] = t;
  }
  __syncthreads();
  const float gmax = red[0];
  __syncthreads();

  // -------- exp + sum reduce --------
  const float e = __expf(v - gmax);
  float s = e;
#pragma unroll
  for (int off = 16; off > 0; off >>= 1) s += __shfl_xor(s, off, 32);
  if (lane == 0) red[wave] = s;
  __syncthreads();
  if (wave == 0) {
    float t = red[lane];
#pragma unroll
    for (int off = 16; off > 0; off >>= 1) t += __shfl_xor(t, off, 32);
    if (lane == 0) red[0] = t;
  }
  __syncthreads();
  const float gsum = red[0];

  attn[b * S_ + tid] = e / gsum;
}

// ---------------------------------------------------------------------------
// Kernel 4: c_vector[b,h] = sum_s attn[b,s] * enc_output[b,s,h]
// attn row cached in LDS; enc_output streamed NT, coalesced over h.
// ---------------------------------------------------------------------------
__global__ __launch_bounds__(256) void k_context(const float* __restrict__ enc_output,
                                                 const float* __restrict__ attn,
                                                 float* __restrict__ cvec) {
  __shared__ float a_sh[S_];
  const int b = blockIdx.x;
  for (int s = threadIdx.x; s < S_; s += 256) a_sh[s] = attn[b * S_ + s];
  __syncthreads();

  const int    h0   = threadIdx.x;
  const float* base = enc_output + (size_t)b * S_ * H_;
  float acc0 = 0.f, acc1 = 0.f;
  for (int s = 0; s < S_; ++s) {
    const float a = a_sh[s];
    acc0 += a * __builtin_nontemporal_load(base + (size_t)s * H_ + h0);
    acc1 += a * __builtin_nontemporal_load(base + (size_t)s * H_ + h0 + 256);
  }
  cvec[b * H_ + h0]       = acc0;
  cvec[b * H_ + h0 + 256] = acc1;
}

// ---------------------------------------------------------------------------
extern "C" void kernel_launch(void* const* d_in, const int* in_sizes, int n_in,
                              void* d_out, int out_size, void* d_ws, size_t ws_size,
                              hipStream_t stream) {
  const float* enc_output = (const float*)d_in[0];  // [B,S,H]
  const float* enc_attn   = (const float*)d_in[1];  // [B,S,U]
  const float* coverage   = (const float*)d_in[2];  // [B,S]
  const float* dec_state  = (const float*)d_in[3];  // [B,D]
  const float* W_dec      = (const float*)d_in[4];  // [D,U]
  const float* b_dec      = (const float*)d_in[5];  // [U]
  const float* W_cov      = (const float*)d_in[6];  // [1,U]
  const float* w_attn     = (const float*)d_in[7];  // [U,1]

  float* attn_out = (float*)d_out;             // B*S  (first output, flattened [B,S,1])
  float* cvec_out = (float*)d_out + B_ * S_;   // B*H  (second output)

  float* dec_attn_ws = (float*)d_ws;              // B*U floats
  float* scores_ws   = dec_attn_ws + B_ * U_;     // B*S floats

  k_dec_gemm<<<dim3(B_ / 16, U_ / 16), 32, 0, stream>>>(dec_state, W_dec, b_dec, dec_attn_ws);
  k_scores<<<(B_ * S_) / 8, 256, 0, stream>>>(enc_attn, coverage, dec_attn_ws, W_cov, w_attn,
                                              scores_ws);
  k_softmax<<<B_, 1024, 0, stream>>>(scores_ws, attn_out);
  k_context<<<B_, 256, 0, stream>>>(enc_output, attn_out, cvec_out);
}